// Multi_Head_Attention_26345329394331
// MI455X (gfx1250) — compile-verified
//
#include <hip/hip_runtime.h>
#include <math.h>

// ---------------------------------------------------------------------------
// MHA, fp32 end-to-end on gfx1250 via V_WMMA_F32_16X16X4_F32.
// Problem is HBM-bound (~330MB moved vs ~103 GFLOP), so fp32 WMMA is the
// right precision: exact accumulate, and still far from the matrix-op ceiling.
// ---------------------------------------------------------------------------

#define DIM   1024
#define HEADS 16
#define HDIM  64
#define BATCH 8
#define SEQ   1024

typedef float v2f __attribute__((ext_vector_type(2)));
typedef float v8f __attribute__((ext_vector_type(8)));

__device__ __forceinline__ v8f wmma_f32(v2f a, v2f b, v8f c) {
  // D = A(16x4) * B(4x16) + C(16x16), fp32, wave32
  return __builtin_amdgcn_wmma_f32_16x16x4_f32(false, a, false, b, (short)0, c,
                                               false, false);
}

// ---------------------------------------------------------------------------
// Kernel 1: qkv = x @ w_qkv^T, scattered into q/k/v as [B*H, N, D]
// Block tile 128(M) x 64(N), 256 threads = 8 waves (4x2), wave tile 32x32.
// ---------------------------------------------------------------------------
#define GS 40   // LDS row stride (32 + 8 pad), 160B = 16B aligned
#define KC 32   // K chunk

__global__ __launch_bounds__(256)
void qkv_gemm_kernel(const float* __restrict__ x, const float* __restrict__ w,
                     float* __restrict__ q, float* __restrict__ k,
                     float* __restrict__ v) {
  __shared__ float As[128 * GS];
  __shared__ float Bs[64 * GS];

  const int tid   = threadIdx.x;
  const int lane  = tid & 31;
  const int wave  = tid >> 5;
  const int waveM = wave & 3;
  const int waveN = wave >> 2;
  const int half  = lane >> 4;
  const int lr    = lane & 15;

  const int row0 = blockIdx.x * 128;
  const int col0 = blockIdx.y * 64;

  v8f acc[2][2];
#pragma unroll
  for (int a = 0; a < 2; ++a)
#pragma unroll
    for (int b = 0; b < 2; ++b)
#pragma unroll
      for (int i = 0; i < 8; ++i) acc[a][b][i] = 0.0f;

  for (int k0 = 0; k0 < DIM; k0 += KC) {
    __syncthreads();
    // stage A: 128x32 = 1024 float4, 4 per thread
#pragma unroll
    for (int i = 0; i < 4; ++i) {
      int slot = tid + i * 256;
      int r = slot >> 3, c4 = slot & 7;
      float4 t = *(const float4*)(x + (size_t)(row0 + r) * DIM + k0 + c4 * 4);
      *(float4*)(&As[r * GS + c4 * 4]) = t;
    }
    // stage B (weights [N,K] row-major): 64x32 = 512 float4, 2 per thread
#pragma unroll
    for (int i = 0; i < 2; ++i) {
      int slot = tid + i * 256;
      int r = slot >> 3, c4 = slot & 7;
      float4 t = *(const float4*)(w + (size_t)(col0 + r) * DIM + k0 + c4 * 4);
      *(float4*)(&Bs[r * GS + c4 * 4]) = t;
    }
    __syncthreads();

#pragma unroll
    for (int kk = 0; kk < KC; kk += 4) {
      v2f a[2], b[2];
#pragma unroll
      for (int i = 0; i < 2; ++i) {
        const float* ap = &As[(waveM * 32 + i * 16 + lr) * GS + kk + half * 2];
        a[i].x = ap[0]; a[i].y = ap[1];
        const float* bp = &Bs[(waveN * 32 + i * 16 + lr) * GS + kk + half * 2];
        b[i].x = bp[0]; b[i].y = bp[1];
      }
      acc[0][0] = wmma_f32(a[0], b[0], acc[0][0]);
      acc[0][1] = wmma_f32(a[0], b[1], acc[0][1]);
      acc[1][0] = wmma_f32(a[1], b[0], acc[1][0]);
      acc[1][1] = wmma_f32(a[1], b[1], acc[1][1]);
    }
  }

  // Epilogue scatter: this 64-wide col tile is exactly one (s, h) pair.
  const int s = col0 >> 10;            // 0=q, 1=k, 2=v
  const int h = (col0 & 1023) >> 6;    // head
  float* dst = (s == 0) ? q : (s == 1) ? k : v;
#pragma unroll
  for (int fm = 0; fm < 2; ++fm)
#pragma unroll
    for (int fn = 0; fn < 2; ++fn)
#pragma unroll
      for (int i = 0; i < 8; ++i) {
        int grow = row0 + waveM * 32 + fm * 16 + half * 8 + i;
        int dcol = waveN * 32 + fn * 16 + lr;   // d within head
        int bb = grow >> 10, n = grow & 1023;
        dst[(((size_t)(bb * HEADS + h) * SEQ) + n) * HDIM + dcol] =
            acc[fm][fn][i];
      }
}

// ---------------------------------------------------------------------------
// Kernel 2: flash attention per (b*h, 64-query tile). 128 threads = 4 waves,
// 16 query rows per wave. Keys streamed in chunks of 32 through LDS.
// ---------------------------------------------------------------------------
#define QT 64       // query rows per block
#define KT 32       // key chunk
#define DS 68       // LDS stride for 64-wide rows (272B, 16B aligned)
#define PS 40       // LDS stride for 32-wide P rows

__global__ __launch_bounds__(128)
void attn_kernel(const float* __restrict__ q, const float* __restrict__ k,
                 const float* __restrict__ v, float* __restrict__ o) {
  __shared__ float Qs[QT * DS];   // 17 KB
  __shared__ float Ks[KT * DS];   // 8.5 KB
  __shared__ float Vs[KT * DS];   // 8.5 KB
  __shared__ float Ps[QT * PS];   // 10 KB (per-wave 16-row strips)

  const int tid  = threadIdx.x;
  const int lane = tid & 31;
  const int wave = tid >> 5;      // 0..3
  const int half = lane >> 4;
  const int lr   = lane & 15;

  const int bh = blockIdx.y;           // b*HEADS + h
  const int q0 = blockIdx.x * QT;

  const float* qp = q + (size_t)bh * SEQ * HDIM;
  const float* kp = k + (size_t)bh * SEQ * HDIM;
  const float* vp = v + (size_t)bh * SEQ * HDIM;

  // Stage Q once, folding in softmax scale 1/sqrt(64).
  const float scale = 0.125f;
#pragma unroll
  for (int i = 0; i < 8; ++i) {
    int slot = tid + i * 128;
    int r = slot >> 4, c4 = slot & 15;
    float4 t = *(const float4*)(qp + (size_t)(q0 + r) * HDIM + c4 * 4);
    t.x *= scale; t.y *= scale; t.z *= scale; t.w *= scale;
    *(float4*)(&Qs[r * DS + c4 * 4]) = t;
  }

  float m_run[8], l_run[8];
  v8f oacc[4];
#pragma unroll
  for (int i = 0; i < 8; ++i) { m_run[i] = -1e30f; l_run[i] = 0.0f; }
#pragma unroll
  for (int j = 0; j < 4; ++j)
#pragma unroll
    for (int i = 0; i < 8; ++i) oacc[j][i] = 0.0f;

  for (int kc = 0; kc < SEQ; kc += KT) {
    __syncthreads();
    // Stage K/V chunk: 32x64 each = 512 float4 each, 4+4 per thread.
#pragma unroll
    for (int i = 0; i < 4; ++i) {
      int slot = tid + i * 128;
      int r = slot >> 4, c4 = slot & 15;
      *(float4*)(&Ks[r * DS + c4 * 4]) =
          *(const float4*)(kp + (size_t)(kc + r) * HDIM + c4 * 4);
      *(float4*)(&Vs[r * DS + c4 * 4]) =
          *(const float4*)(vp + (size_t)(kc + r) * HDIM + c4 * 4);
    }
    __syncthreads();

    // S = Q(16 rows) @ K^T  -> 2 frags (16x16) over D=64
    v8f s[2];
#pragma unroll
    for (int j = 0; j < 2; ++j)
#pragma unroll
      for (int i = 0; i < 8; ++i) s[j][i] = 0.0f;
#pragma unroll
    for (int kk = 0; kk < HDIM; kk += 4) {
      v2f a;
      const float* ap = &Qs[(wave * 16 + lr) * DS + kk + half * 2];
      a.x = ap[0]; a.y = ap[1];
#pragma unroll
      for (int j = 0; j < 2; ++j) {
        v2f b;
        const float* bp = &Ks[(j * 16 + lr) * DS + kk + half * 2];
        b.x = bp[0]; b.y = bp[1];
        s[j] = wmma_f32(a, b, s[j]);
      }
    }

    // Online softmax. Row (wave*16 + half*8 + i) lives in s[*][i] across the
    // 16 lanes of this half; xor masks 1..8 stay inside each half.
#pragma unroll
    for (int i = 0; i < 8; ++i) {
      float mx = fmaxf(s[0][i], s[1][i]);
#pragma unroll
      for (int m = 1; m <= 8; m <<= 1) mx = fmaxf(mx, __shfl_xor(mx, m, 32));
      float mnew = fmaxf(m_run[i], mx);
      float sum = 0.0f;
#pragma unroll
      for (int j = 0; j < 2; ++j) {
        float p = __expf(s[j][i] - mnew);
        s[j][i] = p;
        sum += p;
      }
#pragma unroll
      for (int m = 1; m <= 8; m <<= 1) sum += __shfl_xor(sum, m, 32);
      float alpha = __expf(m_run[i] - mnew);
      l_run[i] = l_run[i] * alpha + sum;
      m_run[i] = mnew;
#pragma unroll
      for (int j = 0; j < 4; ++j) oacc[j][i] *= alpha;
    }

    // P: C-layout frags -> row-major per-wave LDS strip (wave-local, no barrier)
#pragma unroll
    for (int j = 0; j < 2; ++j)
#pragma unroll
      for (int i = 0; i < 8; ++i)
        Ps[(wave * 16 + half * 8 + i) * PS + j * 16 + lr] = s[j][i];

    // O += P(16x32) @ V(32x64) -> 4 frags
#pragma unroll
    for (int kk = 0; kk < KT; kk += 4) {
      v2f a;
      const float* ap = &Ps[(wave * 16 + lr) * PS + kk + half * 2];
      a.x = ap[0]; a.y = ap[1];
#pragma unroll
      for (int j = 0; j < 4; ++j) {
        v2f b;
        b.x = Vs[(kk + half * 2 + 0) * DS + j * 16 + lr];
        b.y = Vs[(kk + half * 2 + 1) * DS + j * 16 + lr];
        oacc[j] = wmma_f32(a, b, oacc[j]);
      }
    }
  }

  // Normalize and store O as [B, N, H*D] so proj GEMM reads it row-major.
  const int bb = bh >> 4, h = bh & 15;
#pragma unroll
  for (int j = 0; j < 4; ++j)
#pragma unroll
    for (int i = 0; i < 8; ++i) {
      int qrow = q0 + wave * 16 + half * 8 + i;
      int d = j * 16 + lr;
      o[((size_t)(bb * SEQ + qrow) * DIM) + h * HDIM + d] =
          oacc[j][i] / l_run[i];
    }
}

// ---------------------------------------------------------------------------
// Kernel 3: out = o @ w_proj^T + b. Same tiling as kernel 1.
// ---------------------------------------------------------------------------
__global__ __launch_bounds__(256)
void proj_gemm_kernel(const float* __restrict__ a_in,
                      const float* __restrict__ w,
                      const float* __restrict__ bias,
                      float* __restrict__ out) {
  __shared__ float As[128 * GS];
  __shared__ float Bs[64 * GS];

  const int tid   = threadIdx.x;
  const int lane  = tid & 31;
  const int wave  = tid >> 5;
  const int waveM = wave & 3;
  const int waveN = wave >> 2;
  const int half  = lane >> 4;
  const int lr    = lane & 15;

  const int row0 = blockIdx.x * 128;
  const int col0 = blockIdx.y * 64;

  v8f acc[2][2];
#pragma unroll
  for (int a = 0; a < 2; ++a)
#pragma unroll
    for (int b = 0; b < 2; ++b)
#pragma unroll
      for (int i = 0; i < 8; ++i) acc[a][b][i] = 0.0f;

  for (int k0 = 0; k0 < DIM; k0 += KC) {
    __syncthreads();
#pragma unroll
    for (int i = 0; i < 4; ++i) {
      int slot = tid + i * 256;
      int r = slot >> 3, c4 = slot & 7;
      *(float4*)(&As[r * GS + c4 * 4]) =
          *(const float4*)(a_in + (size_t)(row0 + r) * DIM + k0 + c4 * 4);
    }
#pragma unroll
    for (int i = 0; i < 2; ++i) {
      int slot = tid + i * 256;
      int r = slot >> 3, c4 = slot & 7;
      *(float4*)(&Bs[r * GS + c4 * 4]) =
          *(const float4*)(w + (size_t)(col0 + r) * DIM + k0 + c4 * 4);
    }
    __syncthreads();

#pragma unroll
    for (int kk = 0; kk < KC; kk += 4) {
      v2f a[2], b[2];
#pragma unroll
      for (int i = 0; i < 2; ++i) {
        const float* ap = &As[(waveM * 32 + i * 16 + lr) * GS + kk + half * 2];
        a[i].x = ap[0]; a[i].y = ap[1];
        const float* bp = &Bs[(waveN * 32 + i * 16 + lr) * GS + kk + half * 2];
        b[i].x = bp[0]; b[i].y = bp[1];
      }
      acc[0][0] = wmma_f32(a[0], b[0], acc[0][0]);
      acc[0][1] = wmma_f32(a[0], b[1], acc[0][1]);
      acc[1][0] = wmma_f32(a[1], b[0], acc[1][0]);
      acc[1][1] = wmma_f32(a[1], b[1], acc[1][1]);
    }
  }

#pragma unroll
  for (int fm = 0; fm < 2; ++fm)
#pragma unroll
    for (int fn = 0; fn < 2; ++fn)
#pragma unroll
      for (int i = 0; i < 8; ++i) {
        int grow = row0 + waveM * 32 + fm * 16 + half * 8 + i;
        int gcol = col0 + waveN * 32 + fn * 16 + lr;
        out[(size_t)grow * DIM + gcol] = acc[fm][fn][i] + bias[gcol];
      }
}

// ---------------------------------------------------------------------------
extern "C" void kernel_launch(void* const* d_in, const int* in_sizes, int n_in,
                              void* d_out, int out_size, void* d_ws,
                              size_t ws_size, hipStream_t stream) {
  const float* x      = (const float*)d_in[0];  // [8,1024,1024]
  const float* w_qkv  = (const float*)d_in[1];  // [3072,1024]
  const float* w_proj = (const float*)d_in[2];  // [1024,1024]
  const float* b_proj = (const float*)d_in[3];  // [1024]
  float* out = (float*)d_out;                   // [8,1024,1024]

  // Workspace: q,k,v each [B*H, N, D] fp32 (32MB each) + attn out (32MB).
  const size_t PER = (size_t)BATCH * HEADS * SEQ * HDIM;  // 8,388,608
  float* ws = (float*)d_ws;
  float* q = ws;
  float* k = ws + PER;
  float* v = ws + 2 * PER;
  float* o = ws + 3 * PER;

  // 1) QKV projection: M=8192, N=3072 -> grid 64 x 48
  qkv_gemm_kernel<<<dim3(64, 48), 256, 0, stream>>>(x, w_qkv, q, k, v);

  // 2) Attention: 16 query tiles x 128 (b*h) blocks
  attn_kernel<<<dim3(SEQ / QT, BATCH * HEADS), 128, 0, stream>>>(q, k, v, o);

  // 3) Output projection: M=8192, N=1024 -> grid 64 x 16
  proj_gemm_kernel<<<dim3(64, 16), 256, 0, stream>>>(o, w_proj, b_proj, out);
}